// GraphAttentionLayer_16277926052602
// MI455X (gfx1250) — compile-verified
//
#include <hip/hip_runtime.h>
#include <math.h>

// ---------------------------------------------------------------------------
// GraphAttentionLayer for MI455X (gfx1250).
// B=4, N=256, D=256, H=8, HD=32.  All fp32 (reference dtype) -> use
// V_WMMA_F32_16X16X4_F32 for every GEMM-shaped stage so the matrix pipes are
// used without losing fp32 precision.
//
// Algebraic refactor of the edge MLP (avoids the 268MB (B,N,N,D) tensor):
//   scores_edge[b,h,i,j] = relu(e_ij @ We1 + be1) . T[b,i,h,:] + q_ih.be2_h
//   T[b,i,h,in]          = sum_d We2[in, h*32+d] * Q[b,i,h*32+d]
// ---------------------------------------------------------------------------

typedef __attribute__((ext_vector_type(2))) float v2f;
typedef __attribute__((ext_vector_type(8))) float v8f;

#define Bdim 4
#define Ndim 256
#define Ddim 256
#define Hdim 8
#define HDdim 32
#define INV_SQRT_HD 0.17677669529663689f

__device__ __forceinline__ v8f wmma_f32_k4(v2f a, v2f b, v8f c) {
    // D(16x16,f32) = A(16x4,f32) * B(4x16,f32) + C
    return __builtin_amdgcn_wmma_f32_16x16x4_f32(
        /*neg_a=*/false, a, /*neg_b=*/false, b,
        /*c_mod=*/(short)0, c, /*reuse_a=*/false, /*reuse_b=*/false);
}

// ---------------------------------------------------------------------------
// K1: Q/K/V projections.  Out[m,n] = X[m,:] @ W[:,n] + bias[n]
// 3 GEMMs of (1024 x 256 x 256).  One 16x16 tile per wave, 8 waves/block.
// ---------------------------------------------------------------------------
__global__ __launch_bounds__(256) void qkv_kernel(
    const float* __restrict__ X,
    const float* __restrict__ Wq, const float* __restrict__ bq,
    const float* __restrict__ Wk, const float* __restrict__ bk,
    const float* __restrict__ Wv, const float* __restrict__ bv,
    float* __restrict__ Qo, float* __restrict__ Ko, float* __restrict__ Vo)
{
    const int lane = threadIdx.x & 31;
    const int wave = threadIdx.x >> 5;
    const int tile = blockIdx.x * 8 + wave;      // 0..3071
    const int mat  = tile >> 10;                 // 0:Q 1:K 2:V
    const int rem  = tile & 1023;
    const int mt   = rem >> 4;                   // 0..63 row tile
    const int nt   = rem & 15;                   // 0..15 col tile
    const float* W    = (mat == 0) ? Wq : (mat == 1) ? Wk : Wv;
    const float* bias = (mat == 0) ? bq : (mat == 1) ? bk : bv;
    float*       Out  = (mat == 0) ? Qo : (mat == 1) ? Ko : Vo;

    const int half = lane >> 4;                  // which K/M half this lane feeds
    const int l16  = lane & 15;
    const float* Arow = X + (mt * 16 + l16) * Ddim;

    v8f acc = {};
    for (int k = 0; k < Ddim; k += 4) {
        v2f a, b;
        a.x = Arow[k + 2 * half + 0];
        a.y = Arow[k + 2 * half + 1];
        b.x = W[(k + 2 * half + 0) * Ddim + nt * 16 + l16];
        b.y = W[(k + 2 * half + 1) * Ddim + nt * 16 + l16];
        acc = wmma_f32_k4(a, b, acc);
    }
    const float bn = bias[nt * 16 + l16];
#pragma unroll
    for (int r = 0; r < 8; ++r)
        Out[(mt * 16 + r + 8 * half) * Ddim + nt * 16 + l16] = acc[r] + bn;
}

// ---------------------------------------------------------------------------
// K2: T[bi,h,in] = sum_{d<32} Q[bi,h*32+d] * We2[in,h*32+d]
// 8 GEMMs of (1024 x 256 x 32).  T stored as (1024, 8, 256).
// ---------------------------------------------------------------------------
__global__ __launch_bounds__(256) void tmat_kernel(
    const float* __restrict__ Q, const float* __restrict__ We2,
    float* __restrict__ T)
{
    const int lane = threadIdx.x & 31;
    const int wave = threadIdx.x >> 5;
    const int tile = blockIdx.x * 8 + wave;      // 0..8191
    const int h    = tile >> 10;                 // 0..7
    const int rem  = tile & 1023;
    const int mt   = rem >> 4;                   // bi tile
    const int nt   = rem & 15;                   // "in" tile
    const int half = lane >> 4, l16 = lane & 15;

    const float* Arow = Q + (mt * 16 + l16) * Ddim + h * HDdim;
    const float* Brow = We2 + (nt * 16 + l16) * Ddim + h * HDdim; // B[k][n] = We2[n, h*32+k]

    v8f acc = {};
    for (int k = 0; k < HDdim; k += 4) {
        v2f a, b;
        a.x = Arow[k + 2 * half + 0];
        a.y = Arow[k + 2 * half + 1];
        b.x = Brow[k + 2 * half + 0];
        b.y = Brow[k + 2 * half + 1];
        acc = wmma_f32_k4(a, b, acc);
    }
#pragma unroll
    for (int r = 0; r < 8; ++r)
        T[((mt * 16 + r + 8 * half) * Hdim + h) * Ddim + nt * 16 + l16] = acc[r];
}

// ---------------------------------------------------------------------------
// K3: S[b,h,i,j] = (Q_ih . K_jh) / sqrt(32).  32 GEMMs (256 x 256 x 32).
// ---------------------------------------------------------------------------
__global__ __launch_bounds__(256) void qk_scores_kernel(
    const float* __restrict__ Q, const float* __restrict__ Km,
    float* __restrict__ S)
{
    const int lane = threadIdx.x & 31;
    const int wave = threadIdx.x >> 5;
    const int tile = blockIdx.x * 8 + wave;      // 0..8191
    const int bh   = tile >> 8;                  // 0..31
    const int rem  = tile & 255;
    const int mt   = rem >> 4, nt = rem & 15;
    const int b    = bh >> 3, h = bh & 7;
    const int half = lane >> 4, l16 = lane & 15;

    const float* Arow = Q  + (b * Ndim + mt * 16 + l16) * Ddim + h * HDdim;
    const float* Brow = Km + (b * Ndim + nt * 16 + l16) * Ddim + h * HDdim; // B[k][n]=K[b,n,h*32+k]

    v8f acc = {};
    for (int k = 0; k < HDdim; k += 4) {
        v2f a, b2;
        a.x  = Arow[k + 2 * half + 0];
        a.y  = Arow[k + 2 * half + 1];
        b2.x = Brow[k + 2 * half + 0];
        b2.y = Brow[k + 2 * half + 1];
        acc = wmma_f32_k4(a, b2, acc);
    }
#pragma unroll
    for (int r = 0; r < 8; ++r)
        S[((b * Hdim + h) * Ndim + mt * 16 + r + 8 * half) * Ndim + nt * 16 + l16] =
            acc[r] * INV_SQRT_HD;
}

// ---------------------------------------------------------------------------
// K4: fused edge-MLP contribution + adjacency mask + softmax over j.
// One workgroup per (b,i); thread t handles column j=t; heads looped.
// We1/be1/T row staged in LDS (broadcast reads, conflict-free).
// ---------------------------------------------------------------------------
__global__ __launch_bounds__(256) void edge_softmax_kernel(
    const float* __restrict__ E, const int* __restrict__ Adj,
    const float* __restrict__ We1, const float* __restrict__ be1,
    const float* __restrict__ be2, const float* __restrict__ Q,
    const float* __restrict__ T, float* __restrict__ S)
{
    __shared__ float sWe1[4 * Ddim];
    __shared__ float sbe1[Ddim];
    __shared__ float sT[Hdim * Ddim];
    __shared__ float sqc[Hdim];
    __shared__ float sred[8];

    const int t = threadIdx.x;
    const int lane = t & 31, wv = t >> 5;
    const int bi = blockIdx.x;                   // 0..1023
    const int b  = bi >> 8, i = bi & 255;

    for (int x = t; x < 4 * Ddim; x += 256) sWe1[x] = We1[x];
    sbe1[t] = be1[t];
    for (int x = t; x < Hdim * Ddim; x += 256) sT[x] = T[bi * (Hdim * Ddim) + x];

    // qc[h] = Q[bi, h*32..h*32+31] . be2 slice  (wave wv == head wv)
    float qv = Q[bi * Ddim + t] * be2[t];
#pragma unroll
    for (int off = 16; off; off >>= 1) qv += __shfl_xor(qv, off);
    if (lane == 0) sqc[wv] = qv;
    __syncthreads();

    const int j = t;
    const float4 e = reinterpret_cast<const float4*>(E)[bi * Ndim + j];

    float acc[Hdim];
#pragma unroll
    for (int h = 0; h < Hdim; ++h) acc[h] = 0.f;

#pragma unroll 4
    for (int in = 0; in < Ddim; ++in) {
        float hv = sbe1[in] + e.x * sWe1[in] + e.y * sWe1[Ddim + in] +
                   e.z * sWe1[2 * Ddim + in] + e.w * sWe1[3 * Ddim + in];
        hv = fmaxf(hv, 0.f);
#pragma unroll
        for (int h = 0; h < Hdim; ++h) acc[h] += hv * sT[h * Ddim + in];
    }

    const bool ok = Adj[bi * Ndim + j] != 0;

#pragma unroll 1
    for (int h = 0; h < Hdim; ++h) {
        const int sidx = ((b * Hdim + h) * Ndim + i) * Ndim + j;
        const float s = S[sidx] + acc[h] + sqc[h];

        float m = ok ? s : -3.402823466e38f;
#pragma unroll
        for (int off = 16; off; off >>= 1) m = fmaxf(m, __shfl_xor(m, off));
        if (lane == 0) sred[wv] = m;
        __syncthreads();
        float gmax = sred[0];
#pragma unroll
        for (int w2 = 1; w2 < 8; ++w2) gmax = fmaxf(gmax, sred[w2]);

        const float ex = ok ? __expf(s - gmax) : 0.f;
        float sum = ex;
#pragma unroll
        for (int off = 16; off; off >>= 1) sum += __shfl_xor(sum, off);
        __syncthreads();                 // sred reuse
        if (lane == 0) sred[wv] = sum;
        __syncthreads();
        float gsum = 0.f;
#pragma unroll
        for (int w2 = 0; w2 < 8; ++w2) gsum += sred[w2];

        S[sidx] = ex / gsum;             // attn written in place
        __syncthreads();                 // before next head reuses sred
    }
}

// ---------------------------------------------------------------------------
// K5: O[b,i,h*32+d] = sum_j attn[b,h,i,j] * V[b,j,h*32+d]
// 32 GEMMs of (256 x 32 x 256).
// ---------------------------------------------------------------------------
__global__ __launch_bounds__(256) void av_kernel(
    const float* __restrict__ S, const float* __restrict__ V,
    float* __restrict__ O)
{
    const int lane = threadIdx.x & 31;
    const int wave = threadIdx.x >> 5;
    const int tile = blockIdx.x * 8 + wave;      // 0..1023
    const int bh   = tile >> 5;                  // 32 tiles per (b,h)
    const int rem  = tile & 31;
    const int mt   = rem >> 1;                   // 0..15 (i tile)
    const int nt   = rem & 1;                    // 0..1  (d tile)
    const int b    = bh >> 3, h = bh & 7;
    const int half = lane >> 4, l16 = lane & 15;

    const float* Arow = S + ((size_t)((b * Hdim + h) * Ndim + mt * 16 + l16)) * Ndim;

    v8f acc = {};
    for (int k = 0; k < Ndim; k += 4) {
        v2f a, bb;
        a.x  = Arow[k + 2 * half + 0];
        a.y  = Arow[k + 2 * half + 1];
        bb.x = V[(b * Ndim + k + 2 * half + 0) * Ddim + h * HDdim + nt * 16 + l16];
        bb.y = V[(b * Ndim + k + 2 * half + 1) * Ddim + h * HDdim + nt * 16 + l16];
        acc = wmma_f32_k4(a, bb, acc);
    }
#pragma unroll
    for (int r = 0; r < 8; ++r)
        O[(b * Ndim + mt * 16 + r + 8 * half) * Ddim + h * HDdim + nt * 16 + l16] = acc[r];
}

// ---------------------------------------------------------------------------
// K6: residual + LayerNorm per (b,i) row of 256.
// ---------------------------------------------------------------------------
__global__ __launch_bounds__(256) void ln_kernel(
    const float* __restrict__ O, const float* __restrict__ X,
    const float* __restrict__ gamma, const float* __restrict__ beta,
    float* __restrict__ out)
{
    __shared__ float parts[8];
    const int t = threadIdx.x, lane = t & 31, wv = t >> 5;
    const int row = blockIdx.x;

    const float x = O[row * Ddim + t] + X[row * Ddim + t];

    float s = x;
#pragma unroll
    for (int off = 16; off; off >>= 1) s += __shfl_xor(s, off);
    if (lane == 0) parts[wv] = s;
    __syncthreads();
    float mean = 0.f;
#pragma unroll
    for (int w = 0; w < 8; ++w) mean += parts[w];
    mean *= (1.f / 256.f);
    __syncthreads();

    const float d = x - mean;
    float v = d * d;
#pragma unroll
    for (int off = 16; off; off >>= 1) v += __shfl_xor(v, off);
    if (lane == 0) parts[wv] = v;
    __syncthreads();
    float var = 0.f;
#pragma unroll
    for (int w = 0; w < 8; ++w) var += parts[w];
    var *= (1.f / 256.f);

    out[row * Ddim + t] = d * rsqrtf(var + 1e-5f) * gamma[t] + beta[t];
}

// ---------------------------------------------------------------------------
extern "C" void kernel_launch(void* const* d_in, const int* in_sizes, int n_in,
                              void* d_out, int out_size, void* d_ws, size_t ws_size,
                              hipStream_t stream)
{
    const float* X    = (const float*)d_in[0];
    const int*   Adj  = (const int*)  d_in[1];
    const float* E    = (const float*)d_in[2];
    const float* Wq   = (const float*)d_in[3];
    const float* bq   = (const float*)d_in[4];
    const float* Wk   = (const float*)d_in[5];
    const float* bk   = (const float*)d_in[6];
    const float* Wv   = (const float*)d_in[7];
    const float* bv   = (const float*)d_in[8];
    const float* We1  = (const float*)d_in[9];
    const float* be1  = (const float*)d_in[10];
    const float* We2  = (const float*)d_in[11];
    const float* be2  = (const float*)d_in[12];
    const float* gam  = (const float*)d_in[13];
    const float* bet  = (const float*)d_in[14];

    float* ws = (float*)d_ws;
    float* Q  = ws;                       // 1024*256
    float* K  = ws + 262144;              // 1024*256
    float* V  = ws + 524288;              // 1024*256
    float* T  = ws + 786432;              // 1024*8*256
    float* S  = ws + 2883584;             // 4*8*256*256 (scores -> attn in place)
    float* O  = ws + 4980736;             // 1024*256    (attn @ V)

    qkv_kernel         <<<384,  256, 0, stream>>>(X, Wq, bq, Wk, bk, Wv, bv, Q, K, V);
    tmat_kernel        <<<1024, 256, 0, stream>>>(Q, We2, T);
    qk_scores_kernel   <<<1024, 256, 0, stream>>>(Q, K, S);
    edge_softmax_kernel<<<1024, 256, 0, stream>>>(E, Adj, We1, be1, be2, Q, T, S);
    av_kernel          <<<128,  256, 0, stream>>>(S, V, O);
    ln_kernel          <<<1024, 256, 0, stream>>>(O, X, gam, bet, (float*)d_out);
}